// SDFChamferLoss_35699768164444
// MI455X (gfx1250) — compile-verified
//
#include <hip/hip_runtime.h>
#include <hip/hip_bf16.h>

// ---------------------------------------------------------------------------
// SDF + Chamfer loss for MI455X (gfx1250, wave32).
//
// Chamfer pairwise term computed as a rank-2 GEMM via V_WMMA_F32_16X16X4_F32:
//   ||p-g||^2 = ||p||^2 + ||g||^2 - 2 * (p . g),   p,g in R^2 (K padded 2->4)
// Mins are taken on squared distances (sqrt is monotone), sqrt applied only to
// the 8064 per-axis minima in the finalize pass.
// ---------------------------------------------------------------------------

typedef __attribute__((ext_vector_type(2))) float v2f;
typedef __attribute__((ext_vector_type(8))) float v8f;

#define HH    64
#define WW    64
#define NPIX  (HH * WW)          // 4096
#define NPTS  8064               // (H-1)*W + H*(W-1)
#define NTILE (NPTS / 16)        // 504
#define BIGF  1e30f
#define EPSF  1e-8f

// ---------------------------------------------------------------------------
// Kernel 1: zero-crossing extraction. One thread per edge, per (img, role).
// Ordering matches reference: vertical edges (63x64, row-major) first,
// then horizontal edges (64x63, row-major).
// role 0 = pred, role 1 = gt. Arrays indexed [(img*2+role)*NPTS + e].
// ---------------------------------------------------------------------------
__global__ __launch_bounds__(256)
void extract_zc_kernel(const float* __restrict__ pred,
                       const float* __restrict__ gt,
                       float* __restrict__ R, float* __restrict__ C,
                       float* __restrict__ M) {
    int t = blockIdx.x * blockDim.x + threadIdx.x;
    if (t >= 2 * 2 * NPTS) return;
    int e    = t % NPTS;
    int role = (t / NPTS) % 2;
    int img  = t / (NPTS * 2);
    const float* sdf = (role == 0 ? pred : gt) + img * NPIX;

    float r, c, v1, v2;
    if (e < (HH - 1) * WW) {                 // vertical edges: 63x64
        int i = e >> 6, j = e & 63;
        v1 = sdf[i * WW + j];
        v2 = sdf[(i + 1) * WW + j];
        float a = fabsf(v1) / (fabsf(v1) + fabsf(v2) + EPSF);
        float frac = (v1 == 0.0f) ? 0.0f : ((v2 == 0.0f) ? 1.0f : a);
        r = (float)i + frac;
        c = (float)j;
    } else {                                  // horizontal edges: 64x63
        int e2 = e - (HH - 1) * WW;
        int i = e2 / (WW - 1), j = e2 % (WW - 1);
        v1 = sdf[i * WW + j];
        v2 = sdf[i * WW + j + 1];
        float a = fabsf(v1) / (fabsf(v1) + fabsf(v2) + EPSF);
        float frac = (v1 == 0.0f) ? 0.0f : ((v2 == 0.0f) ? 1.0f : a);
        r = (float)i;
        c = (float)j + frac;
    }
    bool m = (v1 == 0.0f) || (v2 == 0.0f) || (v1 * v2 < 0.0f);
    int base = (img * 2 + role) * NPTS + e;
    R[base] = r;
    C[base] = c;
    M[base] = m ? 1.0f : 0.0f;
}

// ---------------------------------------------------------------------------
// Kernel 2: per-image L1 sum  Σ|pred - gt|  (divided by NPIX in finalize).
// ---------------------------------------------------------------------------
__global__ __launch_bounds__(256)
void l1_kernel(const float* __restrict__ pred, const float* __restrict__ gt,
               float* __restrict__ L1) {
    int img = blockIdx.x;
    __shared__ float s[256];
    float acc = 0.0f;
    for (int k = threadIdx.x; k < NPIX; k += 256)
        acc += fabsf(pred[img * NPIX + k] - gt[img * NPIX + k]);
    s[threadIdx.x] = acc;
    __syncthreads();
    for (int off = 128; off; off >>= 1) {
        if (threadIdx.x < off) s[threadIdx.x] += s[threadIdx.x + off];
        __syncthreads();
    }
    if (threadIdx.x == 0) L1[img] = s[0];
}

// ---------------------------------------------------------------------------
// Kernel 3: Chamfer row-mins via WMMA. One wave32 per (img, dir, A-tile).
//   dir 0: A = pred pts, B = gt  pts, mask = mg  -> MIN[img][0][i] (m1)
//   dir 1: A = gt  pts, B = pred pts, mask = mp  -> MIN[img][1][j] (m2)
//
// A-frag (16x4 f32, ISA layout): lane<16 holds M=lane, K=0..1; lane>=16 holds
// the zero K=2..3 pad. B-frag symmetric (4x16). Accumulator D[M][N]:
// M = 8*(lane>>4) + vgpr, N = lane&15.
// ---------------------------------------------------------------------------
__global__ __launch_bounds__(32)
void chamfer_min_kernel(const float* __restrict__ R, const float* __restrict__ C,
                        const float* __restrict__ M, float* __restrict__ MIN) {
    int b   = blockIdx.x;
    int ti  = b % NTILE;  b /= NTILE;
    int dir = b % 2;      b /= 2;
    int img = b;

    const int baseA = (img * 2 + dir) * NPTS;        // dir0 -> pred, dir1 -> gt
    const int baseB = (img * 2 + (1 - dir)) * NPTS;  // opposite role

    const int lane = threadIdx.x;
    const int n16  = lane & 15;
    const int hi   = lane >> 4;

    // ---- A tile (loaded once) ----
    float ra = R[baseA + ti * 16 + n16];
    float ca = C[baseA + ti * 16 + n16];
    v2f a;
    a[0] = (hi == 0) ? ra : 0.0f;   // K=0 (row) in lanes 0-15, zero pad K=2
    a[1] = (hi == 0) ? ca : 0.0f;   // K=1 (col) in lanes 0-15, zero pad K=3

    // ||a_m||^2 for the 8 rows this lane's accumulator slice covers.
    __shared__ float sNA[16];
    if (lane < 16) sNA[lane] = ra * ra + ca * ca;
    __syncthreads();
    float na[8];
#pragma unroll
    for (int v = 0; v < 8; ++v) na[v] = sNA[hi * 8 + v];

    float rmin[8];
#pragma unroll
    for (int v = 0; v < 8; ++v) rmin[v] = BIGF;

    for (int tj = 0; tj < NTILE; ++tj) {
        int pb = baseB + tj * 16 + n16;
        float rb = R[pb];
        float cb = C[pb];
        float mb = M[pb];
        if (tj + 1 < NTILE) {                 // prefetch next B tile
            __builtin_prefetch(&R[pb + 16], 0, 0);
            __builtin_prefetch(&C[pb + 16], 0, 0);
            __builtin_prefetch(&M[pb + 16], 0, 0);
        }
        v2f bm;
        bm[0] = (hi == 0) ? rb : 0.0f;
        bm[1] = (hi == 0) ? cb : 0.0f;

        v8f acc = {};
        // D = A(16x4) x B(4x16): dot products p.g for the 16x16 tile
        acc = __builtin_amdgcn_wmma_f32_16x16x4_f32(
            /*neg_a=*/false, a, /*neg_b=*/false, bm,
            /*c_mod=*/(short)0, acc, /*reuse_a=*/false, /*reuse_b=*/false);

        float ng = rb * rb + cb * cb;   // ||b_n||^2 for this lane's column
        bool  ok = (mb > 0.5f);
#pragma unroll
        for (int v = 0; v < 8; ++v) {
            float d2  = fmaxf(na[v] + ng - 2.0f * acc[v], 0.0f);
            float val = ok ? d2 : BIGF;
            rmin[v]   = fminf(rmin[v], val);
        }
    }

    // Min across the 16 lanes sharing the same accumulator row M.
#pragma unroll
    for (int v = 0; v < 8; ++v) {
        float x = rmin[v];
        for (int off = 8; off >= 1; off >>= 1)
            x = fminf(x, __shfl_xor(x, off, 16));
        rmin[v] = x;
    }
    if (n16 == 0) {
        float* out = MIN + (img * 2 + dir) * NPTS + ti * 16 + hi * 8;
#pragma unroll
        for (int v = 0; v < 8; ++v) out[v] = rmin[v];  // still squared dists
    }
}

// ---------------------------------------------------------------------------
// Kernel 4: finalize.  m = sqrt(min_d2) (BIG preserved), masked means,
// cd = -mean1 + mean2, loss_img = L1/NPIX + |cd|, out = mean over images.
// ---------------------------------------------------------------------------
__global__ __launch_bounds__(256)
void finalize_kernel(const float* __restrict__ MIN, const float* __restrict__ M,
                     const float* __restrict__ L1, float* __restrict__ out) {
    __shared__ float s1[256], sp[256], s2[256], sg[256];
    float total = 0.0f;
    for (int img = 0; img < 2; ++img) {
        float a1 = 0.0f, ap = 0.0f, a2 = 0.0f, ag = 0.0f;
        for (int e = threadIdx.x; e < NPTS; e += 256) {
            float mp = M[(img * 2 + 0) * NPTS + e];
            float mg = M[(img * 2 + 1) * NPTS + e];
            float d1 = MIN[(img * 2 + 0) * NPTS + e];
            float d2 = MIN[(img * 2 + 1) * NPTS + e];
            float m1 = (d1 >= 0.5f * BIGF) ? BIGF : sqrtf(d1);
            float m2 = (d2 >= 0.5f * BIGF) ? BIGF : sqrtf(d2);
            a1 += m1 * mp;  ap += mp;
            a2 += m2 * mg;  ag += mg;
        }
        s1[threadIdx.x] = a1; sp[threadIdx.x] = ap;
        s2[threadIdx.x] = a2; sg[threadIdx.x] = ag;
        __syncthreads();
        for (int off = 128; off; off >>= 1) {
            if (threadIdx.x < off) {
                s1[threadIdx.x] += s1[threadIdx.x + off];
                sp[threadIdx.x] += sp[threadIdx.x + off];
                s2[threadIdx.x] += s2[threadIdx.x + off];
                sg[threadIdx.x] += sg[threadIdx.x + off];
            }
            __syncthreads();
        }
        if (threadIdx.x == 0) {
            float mean1 = s1[0] / fmaxf(sp[0], 1.0f);
            float mean2 = s2[0] / fmaxf(sg[0], 1.0f);
            float cd    = -mean1 + mean2;
            total += L1[img] * (1.0f / (float)NPIX) + fabsf(cd);
        }
        __syncthreads();
    }
    if (threadIdx.x == 0) out[0] = 0.5f * total;
}

// ---------------------------------------------------------------------------
// Workspace layout (floats):
//   R   : [0,            4*NPTS)
//   C   : [4*NPTS,       8*NPTS)
//   M   : [8*NPTS,      12*NPTS)
//   MIN : [12*NPTS,     16*NPTS)   (per img: m1 then m2, squared dists)
//   L1  : [16*NPTS,     16*NPTS+2)
// Total ~516 KB. Every cell is fully overwritten each call -> deterministic.
// ---------------------------------------------------------------------------
extern "C" void kernel_launch(void* const* d_in, const int* in_sizes, int n_in,
                              void* d_out, int out_size, void* d_ws, size_t ws_size,
                              hipStream_t stream) {
    const float* pred = (const float*)d_in[0];   // y_pred (2,1,64,64) f32
    const float* gt   = (const float*)d_in[1];   // y_true (2,1,64,64) f32

    float* ws  = (float*)d_ws;
    float* R   = ws;
    float* C   = R + 4 * NPTS;
    float* M   = C + 4 * NPTS;
    float* MIN = M + 4 * NPTS;
    float* L1  = MIN + 4 * NPTS;

    extract_zc_kernel<<<(4 * NPTS + 255) / 256, 256, 0, stream>>>(pred, gt, R, C, M);
    l1_kernel<<<2, 256, 0, stream>>>(pred, gt, L1);
    chamfer_min_kernel<<<2 * 2 * NTILE, 32, 0, stream>>>(R, C, M, MIN);
    finalize_kernel<<<1, 256, 0, stream>>>(MIN, M, L1, (float*)d_out);
}